// DilatedWindowSelfAttention2D_81097572483346
// MI455X (gfx1250) — compile-verified
//
#include <hip/hip_runtime.h>
#include <hip/hip_bf16.h>

typedef float v2f __attribute__((ext_vector_type(2)));
typedef float v8f __attribute__((ext_vector_type(8)));

#define Hh   96
#define Ww   96
#define HW   (96 * 96)
#define Cc   128
#define NPIX (2 * HW)   // B * H * W = 18432

// ---------------------------------------------------------------------------
// Kernel 1: fused Q/K/V 1x1-conv GEMMs via V_WMMA_F32_16X16X4_F32.
//   D[o, p] = sum_c W[o, c] * x[b, c, p]
// One wave computes a 16(o) x 16(p) f32 tile; block of 8 waves covers all
// 128 output channels for one 16-pixel tile. The B-fragment (x) is shared
// by the three WMMA accumulation chains (Wq / Wk / Wv).
// Results stored pixel-major: q[p*128 + o] — convenient for attention gather.
// ---------------------------------------------------------------------------
__global__ void __launch_bounds__(256) qkv_wmma_kernel(
    const float* __restrict__ x,  const float* __restrict__ Wq,
    const float* __restrict__ Wk, const float* __restrict__ Wv,
    float* __restrict__ qo, float* __restrict__ ko, float* __restrict__ vo)
{
    const int lane = threadIdx.x & 31;
    const int wave = threadIdx.x >> 5;        // 0..7 -> 16-channel tile
    const int p0   = blockIdx.x * 16;         // 16 consecutive pixels
    const int b    = p0 / HW;
    const int pix  = p0 - b * HW;             // W=96 % 16 == 0: same row, same batch
    const float* xb = x + (size_t)b * Cc * HW + pix;  // x[b][c][pix+col] = xb[c*HW + col]

    const int o0    = wave * 16;
    const int row   = lane & 15;              // A-matrix M index (o)
    const int khalf = (lane >> 4) << 1;       // lanes 0-15 -> K{0,1}; 16-31 -> K{2,3}
    const int colp  = lane & 15;              // B/D N index (pixel column)

    v8f cq = {}; v8f ck = {}; v8f cv = {};
    #pragma unroll 4
    for (int k0 = 0; k0 < Cc; k0 += 4) {
        v2f bfrag;
        bfrag.x = xb[(size_t)(k0 + khalf) * HW + colp];
        bfrag.y = xb[(size_t)(k0 + khalf + 1) * HW + colp];
        const int widx = (o0 + row) * Cc + k0 + khalf;
        v2f aq, ak, av;
        aq.x = Wq[widx]; aq.y = Wq[widx + 1];
        ak.x = Wk[widx]; ak.y = Wk[widx + 1];
        av.x = Wv[widx]; av.y = Wv[widx + 1];
        cq = __builtin_amdgcn_wmma_f32_16x16x4_f32(false, aq, false, bfrag, (short)0, cq, false, false);
        ck = __builtin_amdgcn_wmma_f32_16x16x4_f32(false, ak, false, bfrag, (short)0, ck, false, false);
        cv = __builtin_amdgcn_wmma_f32_16x16x4_f32(false, av, false, bfrag, (short)0, cv, false, false);
    }

    // D layout: VGPR r -> (M = r + (lane<16 ? 0 : 8), N = lane&15)
    const int rowoff = (lane >> 4) << 3;
    const size_t base = (size_t)(p0 + colp) * Cc + o0 + rowoff;
    #pragma unroll
    for (int r = 0; r < 8; ++r) {
        qo[base + r] = cq[r];
        ko[base + r] = ck[r];
        vo[base + r] = cv[r];
    }
}

// ---------------------------------------------------------------------------
// Kernel 2: dilated 7x7 window attention, one thread per (pixel, head),
// single-pass online softmax. Matches reference mask semantics exactly:
//   in-bounds & mask>0 : logit = dot*scale
//   in-bounds & mask==0: logit = dot*scale - 1e9   (V still contributes)
//   out-of-bounds      : logit = -1e9, V = 0       (still in the denominator)
// ---------------------------------------------------------------------------
__global__ void __launch_bounds__(256) attn_kernel(
    const float* __restrict__ qv, const float* __restrict__ kv,
    const float* __restrict__ vv, const int* __restrict__ msk,
    float* __restrict__ ao)
{
    const int t = blockIdx.x * 256 + threadIdx.x;
    if (t >= NPIX * 4) return;
    const int hd  = t & 3;           // head
    const int p   = t >> 2;          // flat pixel
    const int b   = p / HW;
    const int pix = p - b * HW;
    const int y   = pix / Ww;
    const int x   = pix - y * Ww;

    float q[32];
    const float4* q4 = (const float4*)(qv + (size_t)p * Cc + hd * 32);
    #pragma unroll
    for (int u = 0; u < 8; ++u) {
        float4 t4 = q4[u];
        q[4*u] = t4.x; q[4*u+1] = t4.y; q[4*u+2] = t4.z; q[4*u+3] = t4.w;
    }

    const float scale = 0.17677669529663687f;   // 1/sqrt(32)
    float mx = -1e30f, s = 0.f;
    float acc[32];
    #pragma unroll
    for (int d = 0; d < 32; ++d) acc[d] = 0.f;

    #pragma unroll 1
    for (int wi = 0; wi < 49; ++wi) {
        const int di = wi / 7, dj = wi - di * 7;
        const int ny = y + (di - 3) * 2;
        const int nx = x + (dj - 3) * 2;
        const bool valid = ((unsigned)ny < (unsigned)Hh) && ((unsigned)nx < (unsigned)Ww);

        float logit = -1.0e9f;
        const float4* k4 = nullptr;
        const float4* v4 = nullptr;
        if (valid) {
            const int np = b * HW + ny * Ww + nx;
            k4 = (const float4*)(kv + (size_t)np * Cc + hd * 32);
            v4 = (const float4*)(vv + (size_t)np * Cc + hd * 32);
            float dot = 0.f;
            #pragma unroll
            for (int u = 0; u < 8; ++u) {
                float4 kk = k4[u];
                dot += q[4*u] * kk.x + q[4*u+1] * kk.y + q[4*u+2] * kk.z + q[4*u+3] * kk.w;
            }
            const float mb = (msk[np] > 0) ? 0.f : -1.0e9f;
            logit = dot * scale + mb;
        }

        const float nm   = fmaxf(mx, logit);
        const float corr = __expf(mx - nm);
        const float w    = __expf(logit - nm);
        mx = nm;
        s  = s * corr + w;
        if (valid) {
            #pragma unroll
            for (int u = 0; u < 8; ++u) {
                float4 vvv = v4[u];
                acc[4*u]   = acc[4*u]   * corr + w * vvv.x;
                acc[4*u+1] = acc[4*u+1] * corr + w * vvv.y;
                acc[4*u+2] = acc[4*u+2] * corr + w * vvv.z;
                acc[4*u+3] = acc[4*u+3] * corr + w * vvv.w;
            }
        } else {
            #pragma unroll
            for (int d = 0; d < 32; ++d) acc[d] *= corr;
        }
    }

    const float inv = 1.f / s;
    float* o = ao + (size_t)p * Cc + hd * 32;
    #pragma unroll
    for (int d = 0; d < 32; ++d) o[d] = acc[d] * inv;
}

// ---------------------------------------------------------------------------
// Kernel 3: output projection GEMM via V_WMMA_F32_16X16X4_F32.
//   out[b, o, y, x] = sum_c Wp[o, c] * attout[p, c]
// ---------------------------------------------------------------------------
__global__ void __launch_bounds__(256) proj_wmma_kernel(
    const float* __restrict__ ain, const float* __restrict__ Wp,
    float* __restrict__ out)
{
    const int lane = threadIdx.x & 31;
    const int wave = threadIdx.x >> 5;
    const int p0   = blockIdx.x * 16;
    const int b    = p0 / HW;
    const int pix  = p0 - b * HW;
    const float* aw = ain + (size_t)p0 * Cc;   // attout[(p0+col)*128 + c]

    const int o0    = wave * 16;
    const int row   = lane & 15;
    const int khalf = (lane >> 4) << 1;
    const int colp  = lane & 15;

    v8f c = {};
    #pragma unroll 4
    for (int k0 = 0; k0 < Cc; k0 += 4) {
        v2f a, bfrag;
        const int widx = (o0 + row) * Cc + k0 + khalf;
        a.x = Wp[widx]; a.y = Wp[widx + 1];
        bfrag.x = aw[(size_t)colp * Cc + k0 + khalf];
        bfrag.y = aw[(size_t)colp * Cc + k0 + khalf + 1];
        c = __builtin_amdgcn_wmma_f32_16x16x4_f32(false, a, false, bfrag, (short)0, c, false, false);
    }

    const int rowoff = (lane >> 4) << 3;
    float* dst = out + (size_t)b * Cc * HW + (size_t)(o0 + rowoff) * HW + pix + colp;
    #pragma unroll
    for (int r = 0; r < 8; ++r) dst[(size_t)r * HW] = c[r];
}

// ---------------------------------------------------------------------------
extern "C" void kernel_launch(void* const* d_in, const int* in_sizes, int n_in,
                              void* d_out, int out_size, void* d_ws, size_t ws_size,
                              hipStream_t stream)
{
    const float* x  = (const float*)d_in[0];
    const int*   m  = (const int*)  d_in[1];
    const float* Wq = (const float*)d_in[2];
    const float* Wk = (const float*)d_in[3];
    const float* Wv = (const float*)d_in[4];
    const float* Wp = (const float*)d_in[5];
    float* out = (float*)d_out;

    float* ws = (float*)d_ws;
    const size_t n = (size_t)NPIX * Cc;   // 2,359,296 floats per buffer
    float* qb = ws;
    float* kb = ws + n;
    float* vb = ws + 2 * n;
    float* ab = ws + 3 * n;               // total 37.7 MB of d_ws

    qkv_wmma_kernel<<<NPIX / 16, 256, 0, stream>>>(x, Wq, Wk, Wv, qb, kb, vb);
    attn_kernel<<<(NPIX * 4) / 256, 256, 0, stream>>>(qb, kb, vb, m, ab);
    proj_wmma_kernel<<<NPIX / 16, 256, 0, stream>>>(ab, Wp, out);
}